// SGraphAttention_37108517438298
// MI455X (gfx1250) — compile-verified
//
#include <hip/hip_runtime.h>

typedef __attribute__((ext_vector_type(2))) float v2f;
typedef __attribute__((ext_vector_type(8))) float v8f;

// ---------------------------------------------------------------------------
// Kernel 1: ft = x @ W + b   using V_WMMA_F32_16X16X4_F32 (fp32 matrix path)
//   grid.x = row tiles of 16;  block = 256 threads = 8 waves
//   wave w computes the 16x16 output tile at columns [16w, 16w+16)
// A-fragment layout (16x4 f32): lane l -> row (l&15), K = 2*(l>>4), 2*(l>>4)+1
// B-fragment layout (4x16 f32): lane l -> col (l&15), K = 2*(l>>4), 2*(l>>4)+1
// C/D layout: VGPR g, lane l -> row g + 8*(l>>4), col (l&15)
// ---------------------------------------------------------------------------
__global__ __launch_bounds__(256)
void sgat_gemm_ft(const float* __restrict__ x, const float* __restrict__ W,
                  const float* __restrict__ b, float* __restrict__ ft, int n)
{
    const int lane = threadIdx.x & 31;
    const int wave = threadIdx.x >> 5;      // 0..7 -> column tile
    const int row0 = blockIdx.x * 16;
    const int m    = lane & 15;             // A row within tile / B column
    const int kh   = lane >> 4;             // K-half selector
    const int ncol = wave * 16 + m;         // global output column (0..127)

    int arow = row0 + m;
    if (arow >= n) arow = n - 1;            // clamp reads; stores guarded below
    const float* __restrict__ xrow = x + (size_t)arow * 128;

    v8f acc = {};
    #pragma unroll 4
    for (int k = 0; k < 128; k += 4) {
        v2f av, bv;
        const int kk = k + 2 * kh;
        av.x = xrow[kk + 0];
        av.y = xrow[kk + 1];
        bv.x = W[(size_t)(kk + 0) * 128 + ncol];
        bv.y = W[(size_t)(kk + 1) * 128 + ncol];
        acc = __builtin_amdgcn_wmma_f32_16x16x4_f32(
            /*neg_a=*/false, av, /*neg_b=*/false, bv,
            /*c_mod=*/(short)0, acc, /*reuse_a=*/false, /*reuse_b=*/false);
    }

    const float bias = b[ncol];
    if (row0 + 16 <= n) {
        // Fast path (all full tiles): branch-free, scalar-uniform condition.
        float* __restrict__ base = ft + (size_t)(row0 + 8 * kh) * 128 + ncol;
        #pragma unroll
        for (int g = 0; g < 8; ++g)
            base[(size_t)g * 128] = acc[g] + bias;
    } else {
        // Tail tile only: per-row guard.
        #pragma unroll
        for (int g = 0; g < 8; ++g) {
            const int r = row0 + g + 8 * kh;
            if (r < n) ft[(size_t)r * 128 + ncol] = acc[g] + bias;
        }
    }
}

// ---------------------------------------------------------------------------
// Kernel 2: a1[r] = ft[r,:].attn_l ; a2[r] = ft[r,:].attn_r   (wave per row)
// ---------------------------------------------------------------------------
__global__ __launch_bounds__(256)
void sgat_attn_dot(const float* __restrict__ ft,
                   const float* __restrict__ attn_l,
                   const float* __restrict__ attn_r,
                   float* __restrict__ a1, float* __restrict__ a2, int n)
{
    const int lane = threadIdx.x & 31;
    const int row  = blockIdx.x * (blockDim.x >> 5) + (threadIdx.x >> 5);
    if (row >= n) return;

    const float4 v  = ((const float4*)(ft + (size_t)row * 128))[lane];
    const float4 l4 = ((const float4*)attn_l)[lane];
    const float4 r4 = ((const float4*)attn_r)[lane];

    float s1 = v.x * l4.x + v.y * l4.y + v.z * l4.z + v.w * l4.w;
    float s2 = v.x * r4.x + v.y * r4.y + v.z * r4.z + v.w * r4.w;
    #pragma unroll
    for (int off = 16; off > 0; off >>= 1) {
        s1 += __shfl_xor(s1, off, 32);
        s2 += __shfl_xor(s2, off, 32);
    }
    if (lane == 0) { a1[row] = s1; a2[row] = s2; }
}

// ---------------------------------------------------------------------------
// Kernel 3: hard-concrete gate per edge + segment-sum z[dst] (thread per edge)
//   BETA=0.66, GAMMA=-0.1, ZETA=1.1 -> a = clip(sigmoid(l/0.66)*1.2 - 0.1, 0, 1)
// ---------------------------------------------------------------------------
__global__ __launch_bounds__(256)
void sgat_edge_gate(const float* __restrict__ a1, const float* __restrict__ a2,
                    const float* __restrict__ bias_l0,
                    const int* __restrict__ src, const int* __restrict__ dst,
                    float* __restrict__ gate, float* __restrict__ z, int E)
{
    const int e = blockIdx.x * blockDim.x + threadIdx.x;
    if (e >= E) return;
    const int s = src[e], d = dst[e];
    const float logit = a1[s] + a2[d] + bias_l0[0];
    const float sig = 1.0f / (1.0f + __expf(-logit * (1.0f / 0.66f)));
    float a = sig * 1.2f - 0.1f;
    a = fminf(fmaxf(a, 0.0f), 1.0f);
    if (s == d) a = 1.0f;
    gate[e] = a;
    atomicAdd(&z[d], a);
}

// ---------------------------------------------------------------------------
// Kernel 4: out[dst,:] += (gate/z[dst]) * ft[src,:]   (wave per edge)
//   lane loads float4 (512B/row coalesced), 4 f32 atomics/lane.
//   ft and out are L2-resident (51MB each vs 192MB L2).
// ---------------------------------------------------------------------------
__global__ __launch_bounds__(256)
void sgat_scatter(const float* __restrict__ ft, const float* __restrict__ gate,
                  const float* __restrict__ z,
                  const int* __restrict__ src, const int* __restrict__ dst,
                  float* __restrict__ out, int E)
{
    const int lane = threadIdx.x & 31;
    const int e    = blockIdx.x * (blockDim.x >> 5) + (threadIdx.x >> 5);
    if (e >= E) return;
    const int s = src[e], d = dst[e];
    const float w = gate[e] / z[d];
    const float4 v = ((const float4*)(ft + (size_t)s * 128))[lane];
    float* o = out + (size_t)d * 128 + lane * 4;
    atomicAdd(o + 0, v.x * w);
    atomicAdd(o + 1, v.y * w);
    atomicAdd(o + 2, v.z * w);
    atomicAdd(o + 3, v.w * w);
}

// ---------------------------------------------------------------------------
// Launch: inputs = (x, W, b, attn_l, attn_r, bias_l0, src, dst), all fp32/i32
// ---------------------------------------------------------------------------
extern "C" void kernel_launch(void* const* d_in, const int* in_sizes, int n_in,
                              void* d_out, int out_size, void* d_ws, size_t ws_size,
                              hipStream_t stream)
{
    const float* x       = (const float*)d_in[0];
    const float* W       = (const float*)d_in[1];
    const float* b       = (const float*)d_in[2];
    const float* attn_l  = (const float*)d_in[3];
    const float* attn_r  = (const float*)d_in[4];
    const float* bias_l0 = (const float*)d_in[5];
    const int*   src     = (const int*)d_in[6];
    const int*   dst     = (const int*)d_in[7];
    float*       out     = (float*)d_out;

    const int n = in_sizes[0] / 128;   // nodes
    const int E = in_sizes[6];         // edges (incl. self loops)

    // Workspace layout (floats): ft[n*128] | a1[n] | a2[n] | z[n] | gate[E]
    float* ft   = (float*)d_ws;
    float* a1   = ft + (size_t)n * 128;
    float* a2   = a1 + n;
    float* zbuf = a2 + n;
    float* gate = zbuf + n;

    hipMemsetAsync(zbuf, 0, (size_t)n * sizeof(float), stream);
    hipMemsetAsync(out, 0, (size_t)out_size * sizeof(float), stream);

    sgat_gemm_ft<<<(n + 15) / 16, 256, 0, stream>>>(x, W, b, ft, n);
    sgat_attn_dot<<<(n + 7) / 8, 256, 0, stream>>>(ft, attn_l, attn_r, a1, a2, n);
    sgat_edge_gate<<<(E + 255) / 256, 256, 0, stream>>>(a1, a2, bias_l0, src, dst,
                                                        gate, zbuf, E);
    sgat_scatter<<<(E + 7) / 8, 256, 0, stream>>>(ft, gate, zbuf, src, dst, out, E);
}